// StateSpaceModel_1769526526621
// MI455X (gfx1250) — compile-verified
//
#include <hip/hip_runtime.h>
#include <hip/hip_bf16.h>
#include <stdint.h>

// ---------------------------------------------------------------------------
// SSM forward for MI455X (gfx1250, wave32, WMMA + TDM).
//   Phase 1 (1 block): ZOH discretization via augmented matrix exponential
//     (scaling-and-squaring, no inverse) -> fused weight
//     W(96x96) = [[A_d^T, A_d^T C^T],[B_d^T, B_d^T C^T + D^T]]  -> d_ws
//   Phase 2 (4 blocks x 6 waves): sequential scan. Per step one 16x96x96 GEMM
//     [h_new | y] = [h | u] @ W via 24 chained V_WMMA_F32_16X16X4_F32 with
//     weights resident in VGPRs. u[k] is staged 7 steps ahead by the Tensor
//     Data Mover into an 8-slot LDS ring (TENSORcnt-tracked); the per-step
//     barrier is a hand-rolled LDS-only split barrier so global y stores
//     never stall the recurrence.
// ---------------------------------------------------------------------------

#define SD    64         // state dim
#define ID    32         // input dim
#define OD    32         // output dim
#define BAT   64         // batch
#define SEQL  8192       // sequence length
#define KD    96         // SD + ID  (GEMM K)
#define NDIM  96         // SD + OD  (GEMM N)
#define NSQ   8          // squarings (scale = 2^-8)
#define NTAY  12         // Taylor terms
#define HSTR  68         // h LDS row stride (floats): 4*row bank rotation
#define USTR  36         // u LDS row stride (floats): 36*row mod 64 distinct
#define DEPTH 7          // TDM prefetch distance
#define NSLOT 8          // u ring slots (> DEPTH for WAR safety)

typedef __attribute__((ext_vector_type(2))) float v2f;
typedef __attribute__((ext_vector_type(8))) float v8f;
typedef __attribute__((ext_vector_type(4))) unsigned int u32x4;
typedef __attribute__((ext_vector_type(4))) int i32x4;
typedef __attribute__((ext_vector_type(8))) int i32x8;

// ---------------------------------------------------------------------------
// Kernel 1: discretize + fuse weights. One block, 256 threads.
// ---------------------------------------------------------------------------
__global__ __launch_bounds__(256)
void ssm_precompute(const float* __restrict__ Ag,   // (64,64)
                    const float* __restrict__ Bg,   // (64,32)
                    const float* __restrict__ Cg,   // (32,64)
                    const float* __restrict__ Dg,   // (32,32)
                    float* __restrict__ W)          // (96,96) out
{
    __shared__ float bufA[SD * SD];   // power P
    __shared__ float bufB[SD * SD];   // next power Pn
    __shared__ float bufC[SD * SD];   // E   (-> A_d)
    __shared__ float bufD[SD * SD];   // Phi (later E-square temp)

    float* P   = bufA;
    float* Pn  = bufB;
    float* E   = bufC;
    float* Phi = bufD;

    const int t = threadIdx.x;
    const float scale = 1.0f / 256.0f;   // 2^-NSQ

    for (int i = t; i < SD * SD; i += 256) {
        int r = i >> 6, c = i & 63;
        float v = (r == c) ? 1.0f : 0.0f;
        P[i] = v; E[i] = v; Phi[i] = v;
    }
    __syncthreads();

    // Taylor on A_s = A * 2^-8:  E = sum A_s^k/k!,  Phi = sum A_s^k/(k+1)!
    float fact = 1.0f;
    for (int k = 1; k <= NTAY; k++) {
        fact *= (float)k;
        const float invf  = 1.0f / fact;
        const float invf2 = 1.0f / (fact * (float)(k + 1));
        for (int i = t; i < SD * SD; i += 256) {
            int r = i >> 6, c = i & 63;
            float acc = 0.0f;
            for (int x = 0; x < SD; x++)
                acc += P[r * SD + x] * Ag[x * SD + c];
            Pn[i] = acc * scale;
        }
        __syncthreads();
        for (int i = t; i < SD * SD; i += 256) {
            float p = Pn[i];
            E[i]   += p * invf;
            Phi[i] += p * invf2;
        }
        __syncthreads();
        float* tmp = P; P = Pn; Pn = tmp;
    }

    // F0 = Phi @ (B * 2^-8); reuse power buffers for F / Fn
    float* F  = P;
    float* Fn = Pn;
    for (int i = t; i < SD * ID; i += 256) {
        int r = i >> 5, c = i & 31;
        float acc = 0.0f;
        for (int x = 0; x < SD; x++)
            acc += Phi[r * SD + x] * Bg[x * ID + c];
        F[i] = acc * scale;
    }
    __syncthreads();

    // squaring: (E, F) <- (E@E, E@F + F), NSQ times
    for (int s = 0; s < NSQ; s++) {
        for (int i = t; i < SD * ID; i += 256) {
            int r = i >> 5, c = i & 31;
            float acc = F[i];
            for (int x = 0; x < SD; x++)
                acc += E[r * SD + x] * F[x * ID + c];
            Fn[i] = acc;
        }
        for (int i = t; i < SD * SD; i += 256) {
            int r = i >> 6, c = i & 63;
            float acc = 0.0f;
            for (int x = 0; x < SD; x++)
                acc += E[r * SD + x] * E[x * SD + c];
            Phi[i] = acc;
        }
        __syncthreads();
        float* tmp = E; E = Phi; Phi = tmp;
        tmp = F; F = Fn; Fn = tmp;
    }
    // A_d = E (64x64), B_d = F (64x32).

    // Fused W:  rows 0..63 : [ A_d^T | A_d^T C^T ]
    //           rows 64..95: [ B_d^T | B_d^T C^T + D^T ]
    for (int i = t; i < KD * NDIM; i += 256) {
        int r = i / NDIM, c = i % NDIM;
        float v;
        if (r < SD && c < SD) {
            v = E[c * SD + r];
        } else if (r < SD) {
            int n = c - SD;
            float acc = 0.0f;
            for (int x = 0; x < SD; x++)
                acc += E[x * SD + r] * Cg[n * SD + x];
            v = acc;
        } else if (c < SD) {
            v = F[c * ID + (r - SD)];
        } else {
            int i2 = r - SD, n = c - SD;
            float acc = Dg[n * ID + i2];
            for (int x = 0; x < SD; x++)
                acc += F[x * ID + i2] * Cg[n * SD + x];
            v = acc;
        }
        W[r * NDIM + c] = v;
    }
}

// ---------------------------------------------------------------------------
// TDM: load one 16x32-float u tile (row stride SEQL*ID elements) into LDS
// with a padded row stride of USTR floats (32 data DWORDs + 4 pad DWORDs).
// ---------------------------------------------------------------------------
__device__ __forceinline__ void tdm_load_u_tile(const float* gsrc,
                                                unsigned lds_byte_addr) {
    unsigned long long ga = (unsigned long long)(uintptr_t)gsrc;
    u32x4 g0;
    g0[0] = 1u;                                           // count=1, user desc
    g0[1] = lds_byte_addr;                                // LDS dest (bytes)
    g0[2] = (unsigned)(ga & 0xFFFFFFFFu);                 // global addr lo
    g0[3] = ((unsigned)(ga >> 32) & 0x01FFFFFFu) | (2u << 30); // hi | type=2
    i32x8 g1;
    // data_size=4B(2), pad_enable=1, pad_interval=32 DWORDs(4), pad_amount=4(3)
    g1[0] = (2 << 16) | (1 << 20) | (4 << 22) | (3 << 25);
    g1[1] = (int)(32u << 16);     // tensor_dim0[15:0] = 32  (elements)
    g1[2] = (int)(16u << 16);     // tensor_dim1[15:0] = 16  (rows)
    g1[3] = (int)(32u << 16);     // tile_dim0 = 32
    g1[4] = 16;                   // tile_dim1 = 16
    g1[5] = SEQL * ID;            // tensor_dim0_stride = 262144 elements
    g1[6] = 0;
    g1[7] = 0;
    i32x4 z4 = {0, 0, 0, 0};
#if defined(__clang_major__) && (__clang_major__ >= 23)
    i32x8 z8 = {0, 0, 0, 0, 0, 0, 0, 0};
    __builtin_amdgcn_tensor_load_to_lds(g0, g1, z4, z4, z8, 0);
#else
    __builtin_amdgcn_tensor_load_to_lds(g0, g1, z4, z4, 0);
#endif
}

// LDS-only split barrier: make this wave's DS ops visible, then rendezvous.
// Deliberately does NOT wait on STOREcnt (global y stores drain async).
__device__ __forceinline__ void barrier_lds_only() {
    asm volatile("s_wait_dscnt 0x0\n\t"
                 "s_barrier_signal -1\n\t"
                 "s_barrier_wait -1" ::: "memory");
}

// ---------------------------------------------------------------------------
// Kernel 2: sequential scan. 4 blocks (16 batch rows each) x 192 threads
// (6 waves). Wave n owns output columns [16n,16n+16): n<4 -> h_new tiles,
// n in {4,5} -> y tiles. Wave 5 drives the TDM u pipeline.
// ---------------------------------------------------------------------------
__global__ __launch_bounds__(192)
void ssm_scan(const float* __restrict__ u,    // (BAT, SEQL, ID)
              const float* __restrict__ h0,   // (SD)
              const float* __restrict__ W,    // (96, 96)
              float* __restrict__ y)          // (BAT, SEQL, OD)
{
    __shared__ float H[2][16 * HSTR];       // state ping-pong
    __shared__ float U[NSLOT][16 * USTR];   // u ring, TDM-filled

    const int t    = threadIdx.x;
    const int lane = t & 31;
    const int wave = t >> 5;              // 0..5
    const int bb   = blockIdx.x * 16;     // batch base

    // --- Persistent B fragments of W for this wave's N-tile.
    // B tile 4(K)x16(N): lanes 0-15 hold K=4kb+0,+1; lanes 16-31 K=4kb+2,+3.
    const int nc    = 16 * wave + (lane & 15);
    const int krow0 = (lane >> 4) * 2;
    v2f bfrag[24];
#pragma unroll
    for (int kb = 0; kb < 24; kb++) {
        int kr = 4 * kb + krow0;
        v2f b;
        b.x = W[kr * NDIM + nc];
        b.y = W[(kr + 1) * NDIM + nc];
        bfrag[kb] = b;
    }

    // --- Prologue: TDM-prefetch u[0..DEPTH-1]; init h from h0.
    if (wave == 5) {
        for (int d = 0; d < DEPTH; d++)
            tdm_load_u_tile(u + ((size_t)bb * SEQL + d) * ID,
                            (unsigned)(uintptr_t)&U[d][0]);
        __builtin_amdgcn_s_wait_tensorcnt(DEPTH - 1);   // u[0] resident
    }
    for (int i = t; i < 16 * SD; i += 192) {
        int r = i >> 6, c = i & 63;
        H[0][r * HSTR + c] = h0[c];
    }
    __syncthreads();

    const int arow  = lane & 15;
    const int kofs  = krow0;              // A frag: lanes>=16 take K+2
    const int mbase = (lane >> 4) * 8;    // C/D: lanes>=16 hold rows M+8

    for (int k = 0; k < SEQL; k++) {
        const int cur = k & 1, nxt = cur ^ 1;
        const float* Hc = &H[cur][0];
        const float* Uc = &U[k & (NSLOT - 1)][0];

        v8f acc = {0.f, 0.f, 0.f, 0.f, 0.f, 0.f, 0.f, 0.f};
#pragma unroll
        for (int kb = 0; kb < 16; kb++) {          // K = 0..63 : h part
            v2f a = *(const v2f*)(Hc + arow * HSTR + 4 * kb + kofs);
            acc = __builtin_amdgcn_wmma_f32_16x16x4_f32(
                false, a, false, bfrag[kb], (short)0, acc, false, false);
        }
#pragma unroll
        for (int kb = 16; kb < 24; kb++) {         // K = 64..95 : u part
            v2f a = *(const v2f*)(Uc + arow * USTR + 4 * (kb - 16) + kofs);
            acc = __builtin_amdgcn_wmma_f32_16x16x4_f32(
                false, a, false, bfrag[kb], (short)0, acc, false, false);
        }

        if (wave < 4) {
            // h_new tile -> next-step state buffer
            float* Hn = &H[nxt][0];
            const int col = 16 * wave + (lane & 15);
#pragma unroll
            for (int r = 0; r < 8; r++)
                Hn[(mbase + r) * HSTR + col] = acc[r];
        } else {
            // y tile -> global (b, k, j); drains asynchronously
            const int col = 16 * (wave - 4) + (lane & 15);
#pragma unroll
            for (int r = 0; r < 8; r++)
                y[((size_t)(bb + mbase + r) * SEQL + k) * OD + col] = acc[r];
        }

        // Wave 5: keep the TDM pipeline DEPTH ahead, guarantee u[k+1] landed.
        if (wave == 5) {
            const int kn = k + DEPTH;
            if (kn < SEQL) {
                tdm_load_u_tile(u + ((size_t)bb * SEQL + kn) * ID,
                                (unsigned)(uintptr_t)&U[kn & (NSLOT - 1)][0]);
                __builtin_amdgcn_s_wait_tensorcnt(DEPTH - 1);
            } else {
                __builtin_amdgcn_s_wait_tensorcnt(0);
            }
        }
        barrier_lds_only();
    }
}

// ---------------------------------------------------------------------------
extern "C" void kernel_launch(void* const* d_in, const int* in_sizes, int n_in,
                              void* d_out, int out_size, void* d_ws, size_t ws_size,
                              hipStream_t stream) {
    const float* u  = (const float*)d_in[0];   // (64, 8192, 32)
    const float* A  = (const float*)d_in[1];   // (64, 64)
    const float* B  = (const float*)d_in[2];   // (64, 32)
    const float* C  = (const float*)d_in[3];   // (32, 64)
    const float* D  = (const float*)d_in[4];   // (32, 32)
    const float* h0 = (const float*)d_in[5];   // (64)
    float* y = (float*)d_out;                  // (64, 8192, 32)
    float* W = (float*)d_ws;                   // 96*96 floats = 36 KB

    ssm_precompute<<<1, 256, 0, stream>>>(A, B, C, D, W);
    ssm_scan<<<4, 192, 0, stream>>>(u, h0, W, y);
}